// SO3TransformerLayer_87591563035285
// MI455X (gfx1250) — compile-verified
//
#include <hip/hip_runtime.h>
#include <hip/hip_bf16.h>
#include <math.h>

// ---------------------------------------------------------------------------
// SO3 transformer layer for MI455X (gfx1250, wave32).
// All matrix work on V_WMMA_F32_16X16X4_F32 (exact fp32).
// Per-edge tensor-product weights are never materialized: the radial-MLP
// output h[e,16] is fused with the features via per-edge outer products
// Z[e,(u,k)] = x[e,u]*h[e,k] contracted against a reshaped, shared W3'.
// ---------------------------------------------------------------------------

typedef __attribute__((ext_vector_type(2))) float v2f;
typedef __attribute__((ext_vector_type(8))) float v8f;

#define NNODES 10000
#define NEDGES 128000
#define NODE_TILES 625   /* 10000/16 */
#define EDGE_TILES 8000  /* 128000/16 */

// workspace layout (in floats)
#define OFF_W3P   0
#define OFF_FE    36864
#define OFF_FIJ   (OFF_FE   + NNODES*80)
#define OFF_SHV   (OFF_FIJ  + NEDGES*80)
#define OFF_EXP   (OFF_SHV  + NEDGES*3)
#define OFF_ZSUM  (OFF_EXP  + NEDGES)
#define OFF_AGG   (OFF_ZSUM + NNODES)
#define OFF_FRES  (OFF_AGG  + NNODES*80)
#define OFF_STATS (OFF_FRES + NNODES*80)
#define WS_FLOATS (OFF_STATS + 16)        /* ~50.4 MiB */

// ---------------------------------------------------------------------------

__device__ __forceinline__ v8f v8zero() {
  v8f c;
#pragma unroll
  for (int i = 0; i < 8; i++) c[i] = 0.0f;
  return c;
}

__device__ __forceinline__ v8f wmma4(v2f a, v2f b, v8f c) {
  // D = A(16x4) * B(4x16) + C, fp32
  return __builtin_amdgcn_wmma_f32_16x16x4_f32(false, a, false, b, (short)0, c,
                                               false, false);
}

// GEMM over LDS A[16][LDA] (row-major, padded) and row-major B[K][LDB],
// accumulating a 16x16 tile at column base `nbase` of B.
// rl = lane&15, koff = (lane>>4)*2  (per ISA A/B fragment layout).
template <int K, int LDA, int LDB>
__device__ __forceinline__ v8f gemm_ll(const float* __restrict__ A,
                                       const float* __restrict__ B, int nbase,
                                       v8f c, int rl, int koff) {
#pragma unroll
  for (int kb = 0; kb < K; kb += 4) {
    const int k0 = kb + koff;
    v2f a, b;
    a.x = A[rl * LDA + k0];
    a.y = A[rl * LDA + k0 + 1];
    b.x = B[k0 * LDB + nbase + rl];
    b.y = B[(k0 + 1) * LDB + nbase + rl];
    c = wmma4(a, b, c);
  }
  return c;
}

__device__ __forceinline__ float siluf(float x) { return x / (1.0f + expf(-x)); }
__device__ __forceinline__ float leakyf(float x) { return x > 0.0f ? x : 0.01f * x; }

#define R32 0.17677669529663687f /* 1/sqrt(32) */
#define R16 0.25f                /* 1/sqrt(16) */

// ---------------------------------------------------------------------------
// k_prep: reshape W3 -> W3' (B-layout GEMM operands per path) + zero accums.
//   W3A'[(u*16+k)*32+o] = W3[k][u*32+o]            (K=512, N=32)  off 0
//   W3B'[(u*16+k)*32+o] = W3[k][1024+u*32+o]       (K=256, N=32)  off 16384
//   W3C'[(u*16+k)*16+o] = W3[k][1536+u*16+o]       (K=512, N=16)  off 24576
//   W3D'[(u*16+k)*16+o] = W3[k][2048+u*16+o]       (K=256, N=16)  off 32768
// ---------------------------------------------------------------------------
__global__ void k_prep(const float* __restrict__ W3, float* __restrict__ ws) {
  const int i = blockIdx.x * blockDim.x + threadIdx.x;
  const int stride = gridDim.x * blockDim.x;
  float* w3p = ws + OFF_W3P;
  for (int idx = i; idx < 36864; idx += stride) {
    float v;
    if (idx < 16384) {
      const int o = idx & 31, K = idx >> 5, u = K >> 4, k = K & 15;
      v = W3[k * 2304 + u * 32 + o];
    } else if (idx < 24576) {
      const int j = idx - 16384, o = j & 31, K = j >> 5, u = K >> 4, k = K & 15;
      v = W3[k * 2304 + 1024 + u * 32 + o];
    } else if (idx < 32768) {
      const int j = idx - 24576, o = j & 15, K = j >> 4, u = K >> 4, k = K & 15;
      v = W3[k * 2304 + 1536 + u * 16 + o];
    } else {
      const int j = idx - 32768, o = j & 15, K = j >> 4, u = K >> 4, k = K & 15;
      v = W3[k * 2304 + 2048 + u * 16 + o];
    }
    w3p[idx] = v;
  }
  // zero zsum / agg / fres / stats (atomic accumulators must start at 0)
  float* z = ws + OFF_ZSUM;
  const int nz = WS_FLOATS - OFF_ZSUM;
  for (int idx = i; idx < nz; idx += stride) z[idx] = 0.0f;
}

// ---------------------------------------------------------------------------
// k_node_emb: fe = lin(f; Wnode)  (per-node, 16-node tiles)
// ---------------------------------------------------------------------------
__global__ __launch_bounds__(128) void k_node_emb(const float* __restrict__ f,
                                                  const float* __restrict__ Ws,
                                                  const float* __restrict__ Wv,
                                                  float* __restrict__ fe) {
  __shared__ float WsS[1024], WvS[256];
  __shared__ float sA[4][16 * 33];
  __shared__ float vA[4][3][16 * 17];
  const int t = threadIdx.x;
  for (int i = t; i < 1024; i += 128) WsS[i] = Ws[i];
  for (int i = t; i < 256; i += 128) WvS[i] = Wv[i];
  __syncthreads();
  const int w = t >> 5, lane = t & 31, rl = lane & 15, hi = lane >> 4,
            koff = hi * 2;
  const int tile = blockIdx.x * 4 + w;
  if (tile >= NODE_TILES) return;
  const int n0 = tile * 16;
  for (int i = lane; i < 16 * 80; i += 32) {
    const int n = i / 80, c = i % 80;
    const float x = f[(n0 + n) * 80 + c];
    if (c < 32) sA[w][n * 33 + c] = x;
    else { const int cc = c - 32; vA[w][cc % 3][n * 17 + cc / 3] = x; }
  }
  v8f s0 = gemm_ll<32, 33, 32>(sA[w], WsS, 0, v8zero(), rl, koff);
  v8f s1 = gemm_ll<32, 33, 32>(sA[w], WsS, 16, v8zero(), rl, koff);
  v8f m0 = gemm_ll<16, 17, 16>(vA[w][0], WvS, 0, v8zero(), rl, koff);
  v8f m1 = gemm_ll<16, 17, 16>(vA[w][1], WvS, 0, v8zero(), rl, koff);
  v8f m2 = gemm_ll<16, 17, 16>(vA[w][2], WvS, 0, v8zero(), rl, koff);
#pragma unroll
  for (int r = 0; r < 8; r++) {
    const int n = n0 + hi * 8 + r;
    fe[n * 80 + rl] = s0[r] * R32;
    fe[n * 80 + 16 + rl] = s1[r] * R32;
    fe[n * 80 + 32 + rl * 3 + 0] = m0[r] * R16;
    fe[n * 80 + 32 + rl * 3 + 1] = m1[r] * R16;
    fe[n * 80 + 32 + rl * 3 + 2] = m2[r] * R16;
  }
}

// ---------------------------------------------------------------------------
// k_edge1: geometry + radial MLP + fused Z-GEMM FCTP + lin(Wedge) + logits
// one wave == 16 edges
// ---------------------------------------------------------------------------
__global__ __launch_bounds__(128) void k_edge1(
    const float* __restrict__ fe, const float* __restrict__ pos,
    const int* __restrict__ ei, const float* __restrict__ W1,
    const float* __restrict__ W2, const float* __restrict__ w3p,
    const float* __restrict__ Wes, const float* __restrict__ Wev,
    const float* __restrict__ avec, float* __restrict__ fij,
    float* __restrict__ shvo, float* __restrict__ expz,
    float* __restrict__ zsum) {
  __shared__ float W1s[256], W2s[256], WesS[1024], WevS[256], avS[32];
  __shared__ float esS[4][16 * 33];
  __shared__ float evS[4][3][16 * 17];
  __shared__ float dotS[4][16 * 17];
  __shared__ float bufA[4][16 * 17];
  __shared__ float bufB[4][16 * 17];
  __shared__ float shvS[4][64];
  __shared__ float tsS[4][16 * 33];
  __shared__ float tvS[4][3][16 * 17];
  __shared__ int srcS[4][16], dstS[4][16];

  const int t = threadIdx.x;
  for (int i = t; i < 256; i += 128) { W1s[i] = W1[i]; W2s[i] = W2[i]; WevS[i] = Wev[i]; }
  for (int i = t; i < 1024; i += 128) WesS[i] = Wes[i];
  if (t < 32) avS[t] = avec[t];
  __syncthreads();

  const int w = t >> 5, lane = t & 31, rl = lane & 15, hi = lane >> 4,
            koff = hi * 2;
  const int tile = blockIdx.x * 4 + w;
  const int e0 = tile * 16;

  // geometry: each lane handles edge rl (duplicated in both halves)
  float elen;
  {
    const int e = e0 + rl;
    const int s = ei[e], d = ei[NEDGES + e];
    if (hi == 0) { srcS[w][rl] = s; dstS[w][rl] = d; }
    const float px = pos[s * 3 + 0] - pos[d * 3 + 0];
    const float py = pos[s * 3 + 1] - pos[d * 3 + 1];
    const float pz = pos[s * 3 + 2] - pos[d * 3 + 2];
    elen = sqrtf(px * px + py * py + pz * pz + 1e-12f);
    const float s3 = 1.7320508075688772f / elen;  // sqrt(3)/|r|
    if (hi == 0) {
      shvS[w][rl * 4 + 0] = s3 * px;
      shvS[w][rl * 4 + 1] = s3 * py;
      shvS[w][rl * 4 + 2] = s3 * pz;
      shvo[e * 3 + 0] = s3 * px;
      shvo[e * 3 + 1] = s3 * py;
      shvo[e * 3 + 2] = s3 * pz;
    }
  }

  // smooth-finite radial basis * sqrt(NB)  -> bufA (A-layout [16][17])
  {
    const float step = 3.5f / 17.0f;
    const float fct = 1.14136f * 7.3890560989306495f * 4.0f;  // *e^2*sqrt(16)
#pragma unroll
    for (int j = 0; j < 8; j++) {
      const int b = hi * 8 + j;
      const float vb = 3.5f * (float)(b + 1) / 17.0f;
      const float dd = (elen - vb) / step;
      const float d2 = dd * dd;
      bufA[w][rl * 17 + b] = (d2 < 1.0f) ? fct * expf(-1.0f / (1.0f - d2)) : 0.0f;
    }
  }

  // gather edge features fe[src]+fe[dst] (fe is L2-resident, 3.2MB)
  for (int i = lane; i < 16 * 80; i += 32) {
    const int e = i / 80, c = i % 80;
    const float vsum = fe[srcS[w][e] * 80 + c] + fe[dstS[w][e] * 80 + c];
    if (c < 32) esS[w][e * 33 + c] = vsum;
    else { const int cc = c - 32; evS[w][cc % 3][e * 17 + cc / 3] = vsum; }
  }

  // dot[e][u] = (ev . shv)/sqrt(3)
#pragma unroll
  for (int j = 0; j < 8; j++) {
    const int u = hi * 8 + j;
    const float dv = evS[w][0][rl * 17 + u] * shvS[w][rl * 4 + 0] +
                     evS[w][1][rl * 17 + u] * shvS[w][rl * 4 + 1] +
                     evS[w][2][rl * 17 + u] * shvS[w][rl * 4 + 2];
    dotS[w][rl * 17 + u] = dv * 0.5773502691896258f;
  }

  // radial MLP: h1 = silu(emb@W1/4); h2 = silu(h1@W2/4) -> bufA
  {
    v8f c = gemm_ll<16, 17, 16>(bufA[w], W1s, 0, v8zero(), rl, koff);
#pragma unroll
    for (int r = 0; r < 8; r++) bufB[w][(hi * 8 + r) * 17 + rl] = siluf(c[r] * 0.25f);
    v8f c2 = gemm_ll<16, 17, 16>(bufB[w], W2s, 0, v8zero(), rl, koff);
#pragma unroll
    for (int r = 0; r < 8; r++) bufA[w][(hi * 8 + r) * 17 + rl] = siluf(c2[r] * 0.25f);
  }

  // --- fused FCTP, K=512 group: Z = es (x) h2 against W3A' (2 tiles) + W3C'
  v8f cA0 = v8zero(), cA1 = v8zero(), cC = v8zero();
#pragma unroll 4
  for (int kb = 0; kb < 512; kb += 4) {
    const int K = kb + koff, u = K >> 4, k = K & 15;
    const float esv = esS[w][rl * 33 + u];
    const float h0 = bufA[w][rl * 17 + k], h1 = bufA[w][rl * 17 + k + 1];
    v2f a; a.x = esv * h0; a.y = esv * h1;
    v2f b0, b1, bc;
    b0.x = w3p[K * 32 + rl];            b0.y = w3p[(K + 1) * 32 + rl];
    b1.x = w3p[K * 32 + 16 + rl];       b1.y = w3p[(K + 1) * 32 + 16 + rl];
    bc.x = w3p[24576 + K * 16 + rl];    bc.y = w3p[24576 + (K + 1) * 16 + rl];
    cA0 = wmma4(a, b0, cA0);
    cA1 = wmma4(a, b1, cA1);
    cC = wmma4(a, bc, cC);
  }

  // --- K=256 group: dot (x) h2 vs W3B' (2 tiles); ev_m (x) h2 vs W3D'
  v8f cB0 = v8zero(), cB1 = v8zero(), cD0 = v8zero(), cD1 = v8zero(), cD2 = v8zero();
#pragma unroll 4
  for (int kb = 0; kb < 256; kb += 4) {
    const int K = kb + koff, u = K >> 4, k = K & 15;
    const float h0 = bufA[w][rl * 17 + k], h1 = bufA[w][rl * 17 + k + 1];
    const float dv = dotS[w][rl * 17 + u];
    v2f a; a.x = dv * h0; a.y = dv * h1;
    v2f b0, b1, bd;
    b0.x = w3p[16384 + K * 32 + rl];       b0.y = w3p[16384 + (K + 1) * 32 + rl];
    b1.x = w3p[16384 + K * 32 + 16 + rl];  b1.y = w3p[16384 + (K + 1) * 32 + 16 + rl];
    bd.x = w3p[32768 + K * 16 + rl];       bd.y = w3p[32768 + (K + 1) * 16 + rl];
    cB0 = wmma4(a, b0, cB0);
    cB1 = wmma4(a, b1, cB1);
    const float e0v = evS[w][0][rl * 17 + u];
    const float e1v = evS[w][1][rl * 17 + u];
    const float e2v = evS[w][2][rl * 17 + u];
    v2f a0; a0.x = e0v * h0; a0.y = e0v * h1; cD0 = wmma4(a0, bd, cD0);
    v2f a1; a1.x = e1v * h0; a1.y = e1v * h1; cD1 = wmma4(a1, bd, cD1);
    v2f a2; a2.x = e2v * h0; a2.y = e2v * h1; cD2 = wmma4(a2, bd, cD2);
  }

  // combine into ts/tv (includes the /4 from wtp = h@W3/4)
  {
    const float q1 = 1.0f / 32.0f;            // (1/4)*(1/sqrt(64))
    const float q2 = 0.04419417382415922f;    // (1/4)*(1/sqrt(32))
#pragma unroll
    for (int r = 0; r < 8; r++) {
      const int e = hi * 8 + r;
      tsS[w][e * 33 + rl] = cA0[r] * q1 + cB0[r] * q2;
      tsS[w][e * 33 + 16 + rl] = cA1[r] * q1 + cB1[r] * q2;
      const float cb = cC[r] * q1;
      tvS[w][0][e * 17 + rl] = cb * shvS[w][e * 4 + 0] + cD0[r] * q2;
      tvS[w][1][e * 17 + rl] = cb * shvS[w][e * 4 + 1] + cD1[r] * q2;
      tvS[w][2][e * 17 + rl] = cb * shvS[w][e * 4 + 2] + cD2[r] * q2;
    }
  }

  // f_ij = lin(ts,tv ; Wedge); store + attention logits
  {
    v8f s0 = gemm_ll<32, 33, 32>(tsS[w], WesS, 0, v8zero(), rl, koff);
    v8f s1 = gemm_ll<32, 33, 32>(tsS[w], WesS, 16, v8zero(), rl, koff);
    v8f m0 = gemm_ll<16, 17, 16>(tvS[w][0], WevS, 0, v8zero(), rl, koff);
    v8f m1 = gemm_ll<16, 17, 16>(tvS[w][1], WevS, 0, v8zero(), rl, koff);
    v8f m2 = gemm_ll<16, 17, 16>(tvS[w][2], WevS, 0, v8zero(), rl, koff);
#pragma unroll
    for (int r = 0; r < 8; r++) {
      const int e = e0 + hi * 8 + r;
      const float fs0 = s0[r] * R32, fs1 = s1[r] * R32;
      fij[e * 80 + rl] = fs0;
      fij[e * 80 + 16 + rl] = fs1;
      fij[e * 80 + 32 + rl * 3 + 0] = m0[r] * R16;
      fij[e * 80 + 32 + rl * 3 + 1] = m1[r] * R16;
      fij[e * 80 + 32 + rl * 3 + 2] = m2[r] * R16;
      bufB[w][(hi * 8 + r) * 17 + rl] =
          leakyf(fs0) * avS[rl] + leakyf(fs1) * avS[16 + rl];
    }
  }

  // z = clip(sum, +-10); exp; segment-sum into zsum[dst]
  if (hi == 0) {
    float z = 0.0f;
#pragma unroll
    for (int j = 0; j < 16; j++) z += bufB[w][rl * 17 + j];
    z = fminf(10.0f, fmaxf(-10.0f, z));
    const float ez = expf(z);
    expz[e0 + rl] = ez;
    unsafeAtomicAdd(&zsum[dstS[w][rl]], ez);
  }
}

// ---------------------------------------------------------------------------
// k_edge2: gate + shared-weight FCTP + lin(Wlin) + a_ij-weighted scatter
// ---------------------------------------------------------------------------
__global__ __launch_bounds__(128) void k_edge2(
    const float* __restrict__ fij, const float* __restrict__ shv,
    const float* __restrict__ expz, const float* __restrict__ zsum,
    const int* __restrict__ ei, const float* __restrict__ Wgs,
    const float* __restrict__ Wgv, const float* __restrict__ wnl,
    const float* __restrict__ Wls, const float* __restrict__ Wlv,
    float* __restrict__ agg) {
  __shared__ float WgsS[1536], WgvS[256], WlsS[1024], WlvS[256];
  __shared__ float wAS[1024], wBS[512], wCS[512], wDS[256];
  __shared__ float fsS[4][16 * 33], fvS[4][3][16 * 17];
  __shared__ float msS[4][16 * 33], mvS[4][3][16 * 17], dot2S[4][16 * 17];
  __shared__ float vsS[4][16 * 33], vvS[4][3][16 * 17];
  __shared__ float shvS[4][64], aS[4][16];
  __shared__ int dstS[4][16];

  const int t = threadIdx.x;
  for (int i = t; i < 1536; i += 128) WgsS[i] = Wgs[i];
  for (int i = t; i < 256; i += 128) { WgvS[i] = Wgv[i]; WlvS[i] = Wlv[i]; wDS[i] = wnl[2048 + i]; }
  for (int i = t; i < 1024; i += 128) { WlsS[i] = Wls[i]; wAS[i] = wnl[i]; }
  for (int i = t; i < 512; i += 128) { wBS[i] = wnl[1024 + i]; wCS[i] = wnl[1536 + i]; }
  __syncthreads();

  const int w = t >> 5, lane = t & 31, rl = lane & 15, hi = lane >> 4,
            koff = hi * 2;
  const int tile = blockIdx.x * 4 + w, e0 = tile * 16;

  if (hi == 0) {
    const int e = e0 + rl;
    const int d = ei[NEDGES + e];
    dstS[w][rl] = d;
    const float zz = zsum[d];
    aS[w][rl] = expz[e] / (zz > 0.0f ? zz : 1.0f);
    shvS[w][rl * 4 + 0] = shv[e * 3 + 0];
    shvS[w][rl * 4 + 1] = shv[e * 3 + 1];
    shvS[w][rl * 4 + 2] = shv[e * 3 + 2];
  }

  for (int i = lane; i < 16 * 80; i += 32) {
    const int e = i / 80, c = i % 80;
    const float x = fij[(e0 + e) * 80 + c];
    if (c < 32) fsS[w][e * 33 + c] = x;
    else { const int cc = c - 32; fvS[w][cc % 3][e * 17 + cc / 3] = x; }
  }

  // gate: so = s@Wgate_s/sqrt(32) (48 cols); vo = v@Wgate_v/4; mu = gate(...)
  {
    v8f so0 = gemm_ll<32, 33, 48>(fsS[w], WgsS, 0, v8zero(), rl, koff);
    v8f so1 = gemm_ll<32, 33, 48>(fsS[w], WgsS, 16, v8zero(), rl, koff);
    v8f so2 = gemm_ll<32, 33, 48>(fsS[w], WgsS, 32, v8zero(), rl, koff);
    v8f vo0 = gemm_ll<16, 17, 16>(fvS[w][0], WgvS, 0, v8zero(), rl, koff);
    v8f vo1 = gemm_ll<16, 17, 16>(fvS[w][1], WgvS, 0, v8zero(), rl, koff);
    v8f vo2 = gemm_ll<16, 17, 16>(fvS[w][2], WgvS, 0, v8zero(), rl, koff);
#pragma unroll
    for (int r = 0; r < 8; r++) {
      const int e = hi * 8 + r;
      msS[w][e * 33 + rl] = siluf(so0[r] * R32);
      msS[w][e * 33 + 16 + rl] = siluf(so1[r] * R32);
      const float g = tanhf(so2[r] * R32);
      mvS[w][0][e * 17 + rl] = vo0[r] * 0.25f * g;
      mvS[w][1][e * 17 + rl] = vo1[r] * 0.25f * g;
      mvS[w][2][e * 17 + rl] = vo2[r] * 0.25f * g;
    }
  }

  // dot2[e][u] = (mv . shv)/sqrt(3)
#pragma unroll
  for (int j = 0; j < 8; j++) {
    const int u = hi * 8 + j;
    const float dv = mvS[w][0][rl * 17 + u] * shvS[w][rl * 4 + 0] +
                     mvS[w][1][rl * 17 + u] * shvS[w][rl * 4 + 1] +
                     mvS[w][2][rl * 17 + u] * shvS[w][rl * 4 + 2];
    dot2S[w][rl * 17 + u] = dv * 0.5773502691896258f;
  }

  // shared-weight FCTP (weights_nlmp), shs==1
  {
    const float q8 = 0.125f;
    v8f a0 = gemm_ll<32, 33, 32>(msS[w], wAS, 0, v8zero(), rl, koff);
    v8f a1 = gemm_ll<32, 33, 32>(msS[w], wAS, 16, v8zero(), rl, koff);
    v8f b0 = gemm_ll<16, 17, 32>(dot2S[w], wBS, 0, v8zero(), rl, koff);
    v8f b1 = gemm_ll<16, 17, 32>(dot2S[w], wBS, 16, v8zero(), rl, koff);
    v8f cb = gemm_ll<32, 33, 16>(msS[w], wCS, 0, v8zero(), rl, koff);
    v8f d0 = gemm_ll<16, 17, 16>(mvS[w][0], wDS, 0, v8zero(), rl, koff);
    v8f d1 = gemm_ll<16, 17, 16>(mvS[w][1], wDS, 0, v8zero(), rl, koff);
    v8f d2 = gemm_ll<16, 17, 16>(mvS[w][2], wDS, 0, v8zero(), rl, koff);
#pragma unroll
    for (int r = 0; r < 8; r++) {
      const int e = hi * 8 + r;
      vsS[w][e * 33 + rl] = a0[r] * q8 + b0[r] * R32;
      vsS[w][e * 33 + 16 + rl] = a1[r] * q8 + b1[r] * R32;
      const float c8 = cb[r] * q8;
      vvS[w][0][e * 17 + rl] = c8 * shvS[w][e * 4 + 0] + d0[r] * R32;
      vvS[w][1][e * 17 + rl] = c8 * shvS[w][e * 4 + 1] + d1[r] * R32;
      vvS[w][2][e * 17 + rl] = c8 * shvS[w][e * 4 + 2] + d2[r] * R32;
    }
  }

  // v_ij = lin(vs,vv ; Wlin); scatter a_ij * v_ij into agg[dst]
  {
    v8f s0 = gemm_ll<32, 33, 32>(vsS[w], WlsS, 0, v8zero(), rl, koff);
    v8f s1 = gemm_ll<32, 33, 32>(vsS[w], WlsS, 16, v8zero(), rl, koff);
    v8f m0 = gemm_ll<16, 17, 16>(vvS[w][0], WlvS, 0, v8zero(), rl, koff);
    v8f m1 = gemm_ll<16, 17, 16>(vvS[w][1], WlvS, 0, v8zero(), rl, koff);
    v8f m2 = gemm_ll<16, 17, 16>(vvS[w][2], WlvS, 0, v8zero(), rl, koff);
#pragma unroll
    for (int r = 0; r < 8; r++) {
      const int e = hi * 8 + r;
      const float ae = aS[w][e];
      float* base = &agg[dstS[w][e] * 80];
      unsafeAtomicAdd(base + rl, s0[r] * R32 * ae);
      unsafeAtomicAdd(base + 16 + rl, s1[r] * R32 * ae);
      unsafeAtomicAdd(base + 32 + rl * 3 + 0, m0[r] * R16 * ae);
      unsafeAtomicAdd(base + 32 + rl * 3 + 1, m1[r] * R16 * ae);
      unsafeAtomicAdd(base + 32 + rl * 3 + 2, m2[r] * R16 * ae);
    }
  }
}

// ---------------------------------------------------------------------------
// k_node2a: f_res1 = f + lin(agg; Wen); accumulate LN2 RMS statistics
// ---------------------------------------------------------------------------
__global__ __launch_bounds__(128) void k_node2a(
    const float* __restrict__ f, const float* __restrict__ agg,
    const float* __restrict__ Wens, const float* __restrict__ Wenv,
    float* __restrict__ fres, float* __restrict__ stats) {
  __shared__ float WsS[1024], WvS[256];
  __shared__ float sA[4][16 * 33], vA[4][3][16 * 17];
  const int t = threadIdx.x;
  for (int i = t; i < 1024; i += 128) WsS[i] = Wens[i];
  for (int i = t; i < 256; i += 128) WvS[i] = Wenv[i];
  __syncthreads();
  const int w = t >> 5, lane = t & 31, rl = lane & 15, hi = lane >> 4,
            koff = hi * 2;
  const int tile = blockIdx.x * 4 + w;
  if (tile >= NODE_TILES) return;
  const int n0 = tile * 16;
  for (int i = lane; i < 16 * 80; i += 32) {
    const int n = i / 80, c = i % 80;
    const float x = agg[(n0 + n) * 80 + c];
    if (c < 32) sA[w][n * 33 + c] = x;
    else { const int cc = c - 32; vA[w][cc % 3][n * 17 + cc / 3] = x; }
  }
  v8f s0 = gemm_ll<32, 33, 32>(sA[w], WsS, 0, v8zero(), rl, koff);
  v8f s1 = gemm_ll<32, 33, 32>(sA[w], WsS, 16, v8zero(), rl, koff);
  v8f m0 = gemm_ll<16, 17, 16>(vA[w][0], WvS, 0, v8zero(), rl, koff);
  v8f m1 = gemm_ll<16, 17, 16>(vA[w][1], WvS, 0, v8zero(), rl, koff);
  v8f m2 = gemm_ll<16, 17, 16>(vA[w][2], WvS, 0, v8zero(), rl, koff);
  float ls2 = 0.0f, lv2 = 0.0f;
#pragma unroll
  for (int r = 0; r < 8; r++) {
    const int n = n0 + hi * 8 + r;
    const float a = f[n * 80 + rl] + s0[r] * R32;
    fres[n * 80 + rl] = a; ls2 += a * a;
    const float b = f[n * 80 + 16 + rl] + s1[r] * R32;
    fres[n * 80 + 16 + rl] = b; ls2 += b * b;
    const float c0 = f[n * 80 + 32 + rl * 3 + 0] + m0[r] * R16;
    fres[n * 80 + 32 + rl * 3 + 0] = c0; lv2 += c0 * c0;
    const float c1 = f[n * 80 + 32 + rl * 3 + 1] + m1[r] * R16;
    fres[n * 80 + 32 + rl * 3 + 1] = c1; lv2 += c1 * c1;
    const float c2 = f[n * 80 + 32 + rl * 3 + 2] + m2[r] * R16;
    fres[n * 80 + 32 + rl * 3 + 2] = c2; lv2 += c2 * c2;
  }
  unsafeAtomicAdd(&stats[0], ls2);
  unsafeAtomicAdd(&stats[1], lv2);
}

// ---------------------------------------------------------------------------
// k_node2b: LN2 + gated feed-forward + residual -> output
// ---------------------------------------------------------------------------
__global__ __launch_bounds__(128) void k_node2b(
    const float* __restrict__ fres, const float* __restrict__ stats,
    const float* __restrict__ gsv, const float* __restrict__ bsv,
    const float* __restrict__ gvv, const float* __restrict__ Wf1s,
    const float* __restrict__ Wf1v, const float* __restrict__ Wf2s,
    const float* __restrict__ Wf2v, float* __restrict__ out) {
  __shared__ float Wf1S[1536], Wf1vS[256], Wf2S[1024], Wf2vS[256];
  __shared__ float gsS[32], bsS[32], gvS[16];
  __shared__ float xA[4][16 * 33], xV[4][3][16 * 17];
  __shared__ float gA[4][16 * 33], gV[4][3][16 * 17];
  const int t = threadIdx.x;
  for (int i = t; i < 1536; i += 128) Wf1S[i] = Wf1s[i];
  for (int i = t; i < 256; i += 128) { Wf1vS[i] = Wf1v[i]; Wf2vS[i] = Wf2v[i]; }
  for (int i = t; i < 1024; i += 128) Wf2S[i] = Wf2s[i];
  if (t < 32) { gsS[t] = gsv[t]; bsS[t] = bsv[t]; }
  if (t < 16) gvS[t] = gvv[t];
  __syncthreads();
  const int w = t >> 5, lane = t & 31, rl = lane & 15, hi = lane >> 4,
            koff = hi * 2;
  const int tile = blockIdx.x * 4 + w;
  if (tile >= NODE_TILES) return;
  const int n0 = tile * 16;
  const float invs = rsqrtf(stats[0] / (NNODES * 32.0f) + 1e-6f);
  const float invv = rsqrtf(stats[1] / (NNODES * 16.0f) + 1e-6f);
  for (int i = lane; i < 16 * 80; i += 32) {
    const int n = i / 80, c = i % 80;
    const float x = fres[(n0 + n) * 80 + c];
    if (c < 32) xA[w][n * 33 + c] = x * invs * gsS[c] + bsS[c];
    else {
      const int cc = c - 32, u = cc / 3, m = cc % 3;
      xV[w][m][n * 17 + u] = x * invv * gvS[u];
    }
  }
  // ff layer 1 + gate
  {
    v8f so0 = gemm_ll<32, 33, 48>(xA[w], Wf1S, 0, v8zero(), rl, koff);
    v8f so1 = gemm_ll<32, 33, 48>(xA[w], Wf1S, 16, v8zero(), rl, koff);
    v8f so2 = gemm_ll<32, 33, 48>(xA[w], Wf1S, 32, v8zero(), rl, koff);
    v8f vo0 = gemm_ll<16, 17, 16>(xV[w][0], Wf1vS, 0, v8zero(), rl, koff);
    v8f vo1 = gemm_ll<16, 17, 16>(xV[w][1], Wf1vS, 0, v8zero(), rl, koff);
    v8f vo2 = gemm_ll<16, 17, 16>(xV[w][2], Wf1vS, 0, v8zero(), rl, koff);
#pragma unroll
    for (int r = 0; r < 8; r++) {
      const int e = hi * 8 + r;
      gA[w][e * 33 + rl] = siluf(so0[r] * R32);
      gA[w][e * 33 + 16 + rl] = siluf(so1[r] * R32);
      const float g = tanhf(so2[r] * R32);
      gV[w][0][e * 17 + rl] = vo0[r] * 0.25f * g;
      gV[w][1][e * 17 + rl] = vo1[r] * 0.25f * g;
      gV[w][2][e * 17 + rl] = vo2[r] * 0.25f * g;
    }
  }
  // ff layer 2 (lin) + residual
  {
    v8f s0 = gemm_ll<32, 33, 32>(gA[w], Wf2S, 0, v8zero(), rl, koff);
    v8f s1 = gemm_ll<32, 33, 32>(gA[w], Wf2S, 16, v8zero(), rl, koff);
    v8f m0 = gemm_ll<16, 17, 16>(gV[w][0], Wf2vS, 0, v8zero(), rl, koff);
    v8f m1 = gemm_ll<16, 17, 16>(gV[w][1], Wf2vS, 0, v8zero(), rl, koff);
    v8f m2 = gemm_ll<16, 17, 16>(gV[w][2], Wf2vS, 0, v8zero(), rl, koff);
#pragma unroll
    for (int r = 0; r < 8; r++) {
      const int n = n0 + hi * 8 + r;
      out[n * 80 + rl] = fres[n * 80 + rl] + s0[r] * R32;
      out[n * 80 + 16 + rl] = fres[n * 80 + 16 + rl] + s1[r] * R32;
      out[n * 80 + 32 + rl * 3 + 0] = fres[n * 80 + 32 + rl * 3 + 0] + m0[r] * R16;
      out[n * 80 + 32 + rl * 3 + 1] = fres[n * 80 + 32 + rl * 3 + 1] + m1[r] * R16;
      out[n * 80 + 32 + rl * 3 + 2] = fres[n * 80 + 32 + rl * 3 + 2] + m2[r] * R16;
    }
  }
}

// ---------------------------------------------------------------------------
extern "C" void kernel_launch(void* const* d_in, const int* in_sizes, int n_in,
                              void* d_out, int out_size, void* d_ws,
                              size_t ws_size, hipStream_t stream) {
  const float* f       = (const float*)d_in[0];
  const float* pos     = (const float*)d_in[1];
  const int*   ei      = (const int*)d_in[2];
  // d_in[3..5] = ln1_* (unused by the reference output)
  const float* ln2_gs  = (const float*)d_in[6];
  const float* ln2_bs  = (const float*)d_in[7];
  const float* ln2_gv  = (const float*)d_in[8];
  const float* Wnode_s = (const float*)d_in[9];
  const float* Wnode_v = (const float*)d_in[10];
  const float* W1      = (const float*)d_in[11];
  const float* W2      = (const float*)d_in[12];
  const float* W3      = (const float*)d_in[13];
  const float* wnl     = (const float*)d_in[14];
  const float* Wedge_s = (const float*)d_in[15];
  const float* Wedge_v = (const float*)d_in[16];
  const float* a_vec   = (const float*)d_in[17];
  const float* Wgate_s = (const float*)d_in[18];
  const float* Wgate_v = (const float*)d_in[19];
  const float* Wlin_s  = (const float*)d_in[20];
  const float* Wlin_v  = (const float*)d_in[21];
  const float* Wen_s   = (const float*)d_in[22];
  const float* Wen_v   = (const float*)d_in[23];
  const float* Wff1_s  = (const float*)d_in[24];
  const float* Wff1_v  = (const float*)d_in[25];
  const float* Wff2_s  = (const float*)d_in[26];
  const float* Wff2_v  = (const float*)d_in[27];
  float* ws = (float*)d_ws;
  float* outp = (float*)d_out;

  const int node_blocks = (NODE_TILES + 3) / 4;  // 157
  const int edge_blocks = EDGE_TILES / 4;        // 2000

  k_prep<<<256, 256, 0, stream>>>(W3, ws);
  k_node_emb<<<node_blocks, 128, 0, stream>>>(f, Wnode_s, Wnode_v, ws + OFF_FE);
  k_edge1<<<edge_blocks, 128, 0, stream>>>(
      ws + OFF_FE, pos, ei, W1, W2, ws + OFF_W3P, Wedge_s, Wedge_v, a_vec,
      ws + OFF_FIJ, ws + OFF_SHV, ws + OFF_EXP, ws + OFF_ZSUM);
  k_edge2<<<edge_blocks, 128, 0, stream>>>(
      ws + OFF_FIJ, ws + OFF_SHV, ws + OFF_EXP, ws + OFF_ZSUM, ei, Wgate_s,
      Wgate_v, wnl, Wlin_s, Wlin_v, ws + OFF_AGG);
  k_node2a<<<node_blocks, 128, 0, stream>>>(f, ws + OFF_AGG, Wen_s, Wen_v,
                                            ws + OFF_FRES, ws + OFF_STATS);
  k_node2b<<<node_blocks, 128, 0, stream>>>(ws + OFF_FRES, ws + OFF_STATS,
                                            ln2_gs, ln2_bs, ln2_gv, Wff1_s,
                                            Wff1_v, Wff2_s, Wff2_v, outp);
}